// Voxelization_76828374991760
// MI455X (gfx1250) — compile-verified
//
#include <hip/hip_runtime.h>
#include <cstdint>

// ---------------- problem constants (match reference) ----------------
#define GX 704
#define GY 800
#define GZ 20
#define NUM_CELLS (GX * GY * GZ)    // 11,264,000
#define MAXV 30000
#define MAXP 32
#define SCAN_TILE 2048              // 256 threads x 8 items

// ---------------- init kernels (b128 stores) ----------------
__global__ void k_zero_out(float4* __restrict__ p4, int n4,
                           float* __restrict__ p, int n) {
    for (int i = blockIdx.x * blockDim.x + threadIdx.x; i < n4;
         i += gridDim.x * blockDim.x)
        p4[i] = make_float4(0.f, 0.f, 0.f, 0.f);
    if (blockIdx.x == 0 && threadIdx.x == 0)
        for (int r = n4 * 4; r < n; ++r) p[r] = 0.0f;   // tail (usually empty)
}

__global__ void k_fill_cells(uint4* __restrict__ p4, int n4) {
    int i = blockIdx.x * blockDim.x + threadIdx.x;
    if (i < n4) p4[i] = make_uint4(0xFFFFFFFFu, 0xFFFFFFFFu,
                                   0xFFFFFFFFu, 0xFFFFFFFFu);
}

__global__ void k_init_small(int* __restrict__ cnt, int* __restrict__ fill) {
    int v = blockIdx.x * blockDim.x + threadIdx.x;
    if (v < MAXV) { cnt[v] = 0; fill[v] = 0; }
}

// ---------------- pass 1: cell id per point + first-occurrence min ----------------
__global__ void k_cells(const float4* __restrict__ pts,
                        unsigned int* __restrict__ lin,
                        unsigned int* __restrict__ cellmin, int n) {
    int i = blockIdx.x * blockDim.x + threadIdx.x;
    if (i >= n) return;
    // CDNA5 global_prefetch on the streaming point reads
    if (i + 2048 < n) __builtin_prefetch(&pts[i + 2048], 0, 1);
    float4 p = pts[i];
    // exact float32 replication of floor((p - pmin) / vsize)
    float fx = floorf((p.x - 0.0f)  / 0.1f);
    float fy = floorf((p.y + 40.0f) / 0.1f);
    float fz = floorf((p.z + 3.0f)  / 0.2f);
    bool ok = (fx >= 0.0f) && (fx < (float)GX) &&
              (fy >= 0.0f) && (fy < (float)GY) &&
              (fz >= 0.0f) && (fz < (float)GZ);
    unsigned int L = (unsigned int)NUM_CELLS;  // sentinel for invalid
    if (ok) {
        int cx = (int)fx, cy = (int)fy, cz = (int)fz;
        L = ((unsigned int)cz * GY + (unsigned int)cy) * GX + (unsigned int)cx;
        atomicMin(&cellmin[L], (unsigned int)i);   // order-independent
    }
    lin[i] = L;
}

// ---------------- pass 2: is_first flags + per-block exclusive scan ----------------
__global__ void k_flag_scan(const unsigned int* __restrict__ lin,
                            const unsigned int* __restrict__ cellmin,
                            unsigned char* __restrict__ flags,
                            int* __restrict__ exscan,
                            int* __restrict__ blocksum, int n) {
    const int IT = SCAN_TILE / 256;  // 8 items per thread, blocked layout
    int base = blockIdx.x * SCAN_TILE + threadIdx.x * IT;
    int f[IT];
    int tsum = 0;
    for (int j = 0; j < IT; ++j) {
        int i = base + j;
        int fl = 0;
        if (i < n) {
            unsigned int L = lin[i];
            fl = (L < (unsigned int)NUM_CELLS) &&
                 (cellmin[L] == (unsigned int)i);
            flags[i] = (unsigned char)fl;
        }
        f[j] = fl;
        tsum += fl;
    }
    __shared__ int s[256];
    s[threadIdx.x] = tsum;
    __syncthreads();
    for (int off = 1; off < 256; off <<= 1) {          // Hillis-Steele
        int v = (threadIdx.x >= off) ? s[threadIdx.x - off] : 0;
        __syncthreads();
        s[threadIdx.x] += v;
        __syncthreads();
    }
    int run = s[threadIdx.x] - tsum;                   // thread-exclusive offset
    for (int j = 0; j < IT; ++j) {
        int i = base + j;
        if (i < n) exscan[i] = run;
        run += f[j];
    }
    if (threadIdx.x == 255) blocksum[blockIdx.x] = s[255];
}

__global__ void k_scan_blocks(const int* __restrict__ bs, int* __restrict__ bo,
                              int nb) {
    if (blockIdx.x == 0 && threadIdx.x == 0) {
        int acc = 0;
        for (int b = 0; b < nb; ++b) { bo[b] = acc; acc += bs[b]; }
    }
}

// ---------------- pass 3: vid per cell (overwrite dense table) + coors ----------------
__global__ void k_assign_vid(const unsigned int* __restrict__ lin,
                             const unsigned char* __restrict__ flags,
                             const int* __restrict__ exscan,
                             const int* __restrict__ blockoff,
                             unsigned int* __restrict__ cellmin,
                             float* __restrict__ out_coors, int n) {
    int i = blockIdx.x * blockDim.x + threadIdx.x;
    if (i >= n) return;
    if (!flags[i]) return;
    unsigned int L = lin[i];
    int vid = blockoff[i / SCAN_TILE] + exscan[i];     // order of first appearance
    cellmin[L] = (unsigned int)vid;                    // unique writer per cell
    if (vid < MAXV) {
        int cx = (int)(L % GX);
        int cy = (int)((L / GX) % GY);
        int cz = (int)(L / (GX * GY));
        out_coors[(size_t)vid * 3 + 0] = (float)cz;    // (z, y, x)
        out_coors[(size_t)vid * 3 + 1] = (float)cy;
        out_coors[(size_t)vid * 3 + 2] = (float)cx;
    }
}

// ---------------- pass 4: per-voxel counts ----------------
__global__ void k_count(const unsigned int* __restrict__ lin,
                        const unsigned int* __restrict__ cellmin,
                        int* __restrict__ cnt, int n) {
    int i = blockIdx.x * blockDim.x + threadIdx.x;
    if (i >= n) return;
    unsigned int L = lin[i];
    if (L >= (unsigned int)NUM_CELLS) return;
    unsigned int v = cellmin[L];                       // now holds vid
    if (v < (unsigned int)MAXV) atomicAdd(&cnt[v], 1); // order-independent
}

// ---------------- pass 5: exclusive scan of 30000 counts (one block) ----------------
__global__ void k_scan_cnt(const int* __restrict__ cnt, int* __restrict__ offs) {
    const int CH = (MAXV + 255) / 256;                 // 118 per thread
    int t = threadIdx.x;
    int lo = t * CH;
    int hi = lo + CH; if (hi > MAXV) hi = MAXV;
    int sum = 0;
    for (int i = lo; i < hi; ++i) sum += cnt[i];
    __shared__ int s[256];
    s[t] = sum;
    __syncthreads();
    for (int off = 1; off < 256; off <<= 1) {
        int v = (t >= off) ? s[t - off] : 0;
        __syncthreads();
        s[t] += v;
        __syncthreads();
    }
    int run = s[t] - sum;
    for (int i = lo; i < hi; ++i) { offs[i] = run; run += cnt[i]; }
}

// ---------------- pass 6: scatter point indices into per-voxel buckets ----------------
__global__ void k_scatter(const unsigned int* __restrict__ lin,
                          const unsigned int* __restrict__ cellmin,
                          const int* __restrict__ offs, int* __restrict__ fill,
                          int* __restrict__ plist, int n) {
    int i = blockIdx.x * blockDim.x + threadIdx.x;
    if (i >= n) return;
    unsigned int L = lin[i];
    if (L >= (unsigned int)NUM_CELLS) return;
    unsigned int v = cellmin[L];
    if (v < (unsigned int)MAXV) {
        int pos = offs[v] + atomicAdd(&fill[v], 1);    // set content deterministic
        plist[pos] = i;
    }
}

// ---------------- pass 7: one wave per voxel, emit k smallest indices in order ----------------
__global__ void k_select(const int* __restrict__ offs,
                         const int* __restrict__ cnt,
                         const int* __restrict__ plist,
                         const float4* __restrict__ pts,
                         float4* __restrict__ voxels) {
    int wave = (blockIdx.x * blockDim.x + threadIdx.x) >> 5;   // wave32
    int lane = threadIdx.x & 31;
    if (wave >= MAXV) return;
    int c = cnt[wave];
    if (c == 0) return;
    int base = offs[wave];
    int k = (c < MAXP) ? c : MAXP;
    int thresh = -1;
    for (int s = 0; s < k; ++s) {
        int m = 0x7FFFFFFF;
        for (int p = lane; p < c; p += 32) {           // indices unique per voxel
            int idx = plist[base + p];
            if (idx > thresh && idx < m) m = idx;
        }
        for (int off = 16; off > 0; off >>= 1) {       // butterfly min: all lanes get it
            int o = __shfl_xor(m, off, 32);
            if (o < m) m = o;
        }
        if (lane == 0)
            voxels[(size_t)wave * MAXP + s] = pts[m];  // slot s = s-th smallest index
        thresh = m;
    }
}

// ---------------- pass 8: num_points = min(count, 32) as float ----------------
__global__ void k_numpoints(const int* __restrict__ cnt,
                            float* __restrict__ out_np) {
    int v = blockIdx.x * blockDim.x + threadIdx.x;
    if (v < MAXV) {
        int c = cnt[v];
        out_np[v] = (float)(c < MAXP ? c : MAXP);
    }
}

// ---------------- launcher ----------------
extern "C" void kernel_launch(void* const* d_in, const int* in_sizes, int n_in,
                              void* d_out, int out_size, void* d_ws,
                              size_t ws_size, hipStream_t stream) {
    (void)n_in; (void)ws_size;
    const int N = in_sizes[0] / 4;                     // points: (N, 4) float32
    const float4* pts = (const float4*)d_in[0];

    float* out = (float*)d_out;
    float4* voxels = (float4*)out;                                   // MAXV*MAXP float4
    float* out_coors = out + (size_t)MAXV * MAXP * 4;                // 3,840,000
    float* out_np    = out_coors + (size_t)MAXV * 3;                 // +90,000

    // workspace carve-up (~61 MB worst case); base 256B aligned -> b128 safe
    size_t off = 0;
    char* ws = (char*)d_ws;
    auto take = [&](size_t bytes) -> void* {
        void* p = ws + off;
        off += (bytes + 255) & ~(size_t)255;
        return p;
    };
    const int NB = (N + SCAN_TILE - 1) / SCAN_TILE;
    unsigned int*  cellmin  = (unsigned int*) take((size_t)(NUM_CELLS + 4) * 4);
    unsigned int*  lin      = (unsigned int*) take((size_t)N * 4);
    unsigned char* flags    = (unsigned char*)take((size_t)N);
    int*           exscan   = (int*)          take((size_t)N * 4);
    int*           blocksum = (int*)          take((size_t)NB * 4);
    int*           blockoff = (int*)          take((size_t)NB * 4);
    int*           cnt      = (int*)          take((size_t)MAXV * 4);
    int*           offs     = (int*)          take((size_t)MAXV * 4);
    int*           fill     = (int*)          take((size_t)MAXV * 4);
    int*           plist    = (int*)          take((size_t)N * 4);

    const int T = 256;
    const int gN    = (N + T - 1) / T;
    const int gOut  = 1024;                            // grid-stride b128 zero
    const int nCellFill = (NUM_CELLS + 1 + 3) / 4;     // uint4 count (alloc padded)
    const int gCell = (nCellFill + T - 1) / T;
    const int gV    = (MAXV + T - 1) / T;
    const int gSel  = (MAXV * 32 + T - 1) / T;         // one wave32 per voxel

    // init (b128 stores)
    k_zero_out<<<gOut, T, 0, stream>>>((float4*)out, out_size / 4, out, out_size);
    k_fill_cells<<<gCell, T, 0, stream>>>((uint4*)cellmin, nCellFill);
    k_init_small<<<gV, T, 0, stream>>>(cnt, fill);

    // 1) cell ids + first-occurrence (atomicMin: order-independent)
    k_cells<<<gN, T, 0, stream>>>(pts, lin, cellmin, N);

    // 2) is_first flags + exclusive scan -> vid in order of first appearance
    k_flag_scan<<<NB, T, 0, stream>>>(lin, cellmin, flags, exscan, blocksum, N);
    k_scan_blocks<<<1, 1, 0, stream>>>(blocksum, blockoff, NB);
    k_assign_vid<<<gN, T, 0, stream>>>(lin, flags, exscan, blockoff, cellmin,
                                       out_coors, N);

    // 3) bucket points by voxel: count -> scan -> scatter
    k_count<<<gN, T, 0, stream>>>(lin, cellmin, cnt, N);
    k_scan_cnt<<<1, T, 0, stream>>>(cnt, offs);
    k_scatter<<<gN, T, 0, stream>>>(lin, cellmin, offs, fill, plist, N);

    // 4) per-voxel wave32 selection: slot s = s-th smallest point index
    k_select<<<gSel, T, 0, stream>>>(offs, cnt, plist, pts, voxels);

    // 5) num_points_per_voxel
    k_numpoints<<<gV, T, 0, stream>>>(cnt, out_np);
}